// GraphNormalization_79577154060734
// MI455X (gfx1250) — compile-verified
//
#include <hip/hip_runtime.h>
#include <math.h>

#define DFEAT 128
#define DVEC  (DFEAT / 4)   // 32 float4 per row
#define EPS_N 1e-5f

typedef __attribute__((ext_vector_type(2))) float v2f;
typedef __attribute__((ext_vector_type(4))) float v4f;
typedef __attribute__((ext_vector_type(8))) float v8f;

__device__ __forceinline__ int lower_bound_dev(const int* __restrict__ seg, int n, int key) {
    int lo = 0, hi = n;
    while (lo < hi) {
        int mid = (lo + hi) >> 1;
        if (seg[mid] < key) lo = mid + 1; else hi = mid;
    }
    return lo;
}

// One block per (tensor, segment). 256 threads = 8 wave32.
// Wave w handles row-group w (rows start+w, start+w+8, ...), lane l handles
// float4 column chunk l (columns 4l..4l+3). Pass 1 accumulates sum/sumsq per
// column into registers -> LDS partials -> WMMA ones-vector reduction.
// Pass 2 re-reads (L2-resident) and writes normalized output (NT stores).
__global__ __launch_bounds__(256) void graphnorm_seg_kernel(
    const float* __restrict__ node_feat, const float* __restrict__ edge_feat,
    const float* __restrict__ node_gamma, const float* __restrict__ node_beta,
    const float* __restrict__ edge_gamma, const float* __restrict__ edge_beta,
    const int* __restrict__ node_seg, const int* __restrict__ edge_seg,
    const int* __restrict__ num_graphs_p, int N,
    float* __restrict__ out)
{
    __shared__ float lds_s[8][DFEAT];   // per-rowgroup partial sums
    __shared__ float lds_q[8][DFEAT];   // per-rowgroup partial sumsq
    __shared__ float lds_mean[DFEAT];
    __shared__ float lds_rstd[DFEAT];

    const int B = *num_graphs_p;
    const int tid  = (int)threadIdx.x;
    const int lane = tid & 31;   // wave32 lane; also float4-column index in pass loops
    const int wave = tid >> 5;   // 0..7; also row-group index
    const int c4   = lane;       // float4 column chunk 0..31

    for (int task = (int)blockIdx.x; task < 2 * B; task += (int)gridDim.x) {
        const bool isNode = (task < B);
        const int  sidx   = isNode ? task : (task - B);
        const float* __restrict__ X     = isNode ? node_feat  : edge_feat;
        const int*   __restrict__ seg   = isNode ? node_seg   : edge_seg;
        const float* __restrict__ gamma = isNode ? node_gamma : edge_gamma;
        const float* __restrict__ beta  = isNode ? node_beta  : edge_beta;
        float* __restrict__ Y = out + (isNode ? (size_t)0 : (size_t)N * DFEAT);

        const int start = lower_bound_dev(seg, N, sidx);
        const int end   = lower_bound_dev(seg, N, sidx + 1);
        const int cnt   = end - start;

        // ---------------- Pass 1: per-column sum / sumsq ----------------
        const v4f* __restrict__ X4 = (const v4f*)X;
        v4f acc_s = {0.f, 0.f, 0.f, 0.f};
        v4f acc_q = {0.f, 0.f, 0.f, 0.f};
        #pragma unroll 4
        for (int r = start + wave; r < end; r += 8) {
            v4f v = X4[(size_t)r * DVEC + c4];
            acc_s += v;
            acc_q += v * v;
        }
        {
            const int c0 = 4 * c4;
            lds_s[wave][c0 + 0] = acc_s.x; lds_s[wave][c0 + 1] = acc_s.y;
            lds_s[wave][c0 + 2] = acc_s.z; lds_s[wave][c0 + 3] = acc_s.w;
            lds_q[wave][c0 + 0] = acc_q.x; lds_q[wave][c0 + 1] = acc_q.y;
            lds_q[wave][c0 + 2] = acc_q.z; lds_q[wave][c0 + 3] = acc_q.w;
        }
        __syncthreads();

        // ------- WMMA reduction over the 8 row-groups (ones^T * P) -------
        // Wave w reduces columns [16w, 16w+16). A = ones(16x4).
        // B layout (4x16 f32): lane n<16 -> col n, VGPR0=K0,VGPR1=K1;
        //                      lane n+16 -> col n, VGPR0=K2,VGPR1=K3.
        {
            const int n   = lane & 15;
            const int k0  = (lane >> 4) << 1;     // 0 or 2
            const int col = wave * 16 + n;

            v2f a; a.x = 1.0f; a.y = 1.0f;
            v2f b0s, b1s, b0q, b1q;
            b0s.x = lds_s[k0    ][col]; b0s.y = lds_s[k0 + 1][col];
            b1s.x = lds_s[k0 + 4][col]; b1s.y = lds_s[k0 + 5][col];
            b0q.x = lds_q[k0    ][col]; b0q.y = lds_q[k0 + 1][col];
            b1q.x = lds_q[k0 + 4][col]; b1q.y = lds_q[k0 + 5][col];

            v8f cs = {};
            cs = __builtin_amdgcn_wmma_f32_16x16x4_f32(false, a, false, b0s,
                                                       (short)0, cs, false, false);
            cs = __builtin_amdgcn_wmma_f32_16x16x4_f32(false, a, false, b1s,
                                                       (short)0, cs, false, false);
            v8f cq = {};
            cq = __builtin_amdgcn_wmma_f32_16x16x4_f32(false, a, false, b0q,
                                                       (short)0, cq, false, false);
            cq = __builtin_amdgcn_wmma_f32_16x16x4_f32(false, a, false, b1q,
                                                       (short)0, cq, false, false);

            // D row 0 (VGPR0, lanes 0-15) = column totals.
            if (lane < 16) {
                const float inv  = (cnt > 0) ? (1.0f / (float)cnt) : 0.0f;
                const float mean = cs[0] * inv;
                const float var  = fmaxf(cq[0] * inv - mean * mean, 0.0f);
                lds_mean[col] = mean;
                lds_rstd[col] = 1.0f / (sqrtf(var) + EPS_N);
            }
        }
        __syncthreads();

        // ---------------- Pass 2: normalize + affine (L2 hits) ----------------
        {
            const v4f gm = ((const v4f*)gamma)[c4];
            const v4f bt = ((const v4f*)beta)[c4];
            v4f* __restrict__ Y4 = (v4f*)Y;

            if (cnt > 1) {
                const v4f mn = ((const v4f*)lds_mean)[c4];
                const v4f rs = ((const v4f*)lds_rstd)[c4];
                #pragma unroll 2
                for (int r = start + wave; r < end; r += 8) {
                    v4f v = __builtin_nontemporal_load(&X4[(size_t)r * DVEC + c4]);
                    v4f y = (v - mn) * rs * gm + bt;
                    __builtin_nontemporal_store(y, &Y4[(size_t)r * DVEC + c4]);
                }
            } else {
                // single-row (or empty) segments: passthrough, then affine
                for (int r = start + wave; r < end; r += 8) {
                    v4f v = __builtin_nontemporal_load(&X4[(size_t)r * DVEC + c4]);
                    v4f y = v * gm + bt;
                    __builtin_nontemporal_store(y, &Y4[(size_t)r * DVEC + c4]);
                }
            }
        }
        __syncthreads();   // protect LDS reuse by next grid-stride task
    }
}

extern "C" void kernel_launch(void* const* d_in, const int* in_sizes, int n_in,
                              void* d_out, int out_size, void* d_ws, size_t ws_size,
                              hipStream_t stream) {
    (void)n_in; (void)d_ws; (void)ws_size; (void)out_size;

    const float* node_feat  = (const float*)d_in[0];
    const float* edge_feat  = (const float*)d_in[1];
    const float* node_gamma = (const float*)d_in[2];
    const float* node_beta  = (const float*)d_in[3];
    const float* edge_gamma = (const float*)d_in[4];
    const float* edge_beta  = (const float*)d_in[5];
    const int*   node_seg   = (const int*)d_in[6];
    const int*   edge_seg   = (const int*)d_in[7];
    const int*   num_graphs = (const int*)d_in[8];

    const int D = in_sizes[2];          // feature dim (128 per reference)
    const int N = in_sizes[0] / D;      // rows per tensor

    // 2*B tasks expected (B=1024 -> 2048); grid-stride covers any B read on
    // device. 2048 blocks x 256 threads saturates the GPU.
    const int blocks = 2048;
    graphnorm_seg_kernel<<<blocks, 256, 0, stream>>>(
        node_feat, edge_feat, node_gamma, node_beta, edge_gamma, edge_beta,
        node_seg, edge_seg, num_graphs, N, (float*)d_out);
}